// GraphFusion_5695126634839
// MI455X (gfx1250) — compile-verified
//
#include <hip/hip_runtime.h>
#include <hip/hip_bf16.h>

typedef __attribute__((ext_vector_type(2))) float v2f;
typedef __attribute__((ext_vector_type(8))) float v8f;

#define BB     2048
#define SS     32
#define DD     512
#define TILE_N 1024                 // 32*32 per batch
#define TOTAL  (2048u * 1024u)      // B*S*S elements

// ---------------------------------------------------------------------------
// Kernel 1: per batch, fused L2-normalize + Gram (F F^T) + ReLU via f32 WMMA.
// One block per batch, 256 threads = 8 waves = 8 output 16x16 tiles
// (wave 0..3 -> a_v tiles, wave 4..7 -> a_t tiles).
// ---------------------------------------------------------------------------
__global__ void __launch_bounds__(256)
gram_wmma_kernel(const float* __restrict__ fv, const float* __restrict__ ft,
                 float* __restrict__ av, float* __restrict__ at)
{
    const int b    = blockIdx.x;
    const int tid  = threadIdx.x;
    const int wave = tid >> 5;
    const int lane = tid & 31;

    __shared__ float s_scale[64];   // [0..31]: fv row scales, [32..63]: ft row scales

    const float* __restrict__ Fv = fv + (size_t)b * SS * DD;
    const float* __restrict__ Ft = ft + (size_t)b * SS * DD;

    // ---- row norms: wave w handles rows w, w+8, ... (64 rows total) ----
    for (int r = wave; r < 64; r += 8) {
        const float* src = (r < 32) ? (Fv + (size_t)r * DD)
                                    : (Ft + (size_t)(r - 32) * DD);
        float s = 0.f;
        #pragma unroll 4
        for (int k = lane; k < DD; k += 32) {
            float x = src[k];
            s = fmaf(x, x, s);
        }
        #pragma unroll
        for (int m = 16; m >= 1; m >>= 1) s += __shfl_xor(s, m, 32);
        if (lane == 0) {
            s_scale[r] = 1.0f / fmaxf(sqrtf(s), 1e-12f);
        }
    }
    __syncthreads();

    // ---- WMMA Gram: wave -> (tensor, tile) ----
    const int which = wave >> 2;           // 0 = fv, 1 = ft
    const int ti    = (wave >> 1) & 1;     // row tile
    const int tj    = wave & 1;            // col tile
    const float* __restrict__ F = which ? Ft : Fv;
    float* __restrict__ O       = which ? at : av;
    const float* scl            = s_scale + which * 32;

    // A-fragment (16x4 f32): lane L -> row L%16, K = k0 + 2*(L/16) + {0,1}
    // B-fragment (4x16, = F^T slab): identical per-lane addressing, other slab.
    const int mrow = ti * 16 + (lane & 15);
    const int nrow = tj * 16 + (lane & 15);
    const int koff = (lane >> 4) * 2;

    const float* Arow = F + (size_t)mrow * DD + koff;
    const float* Brow = F + (size_t)nrow * DD + koff;

    v8f acc = {};
    #pragma unroll 8
    for (int k0 = 0; k0 < DD; k0 += 4) {
        v2f a = *(const v2f*)(Arow + k0);
        v2f bm = *(const v2f*)(Brow + k0);
        // 8 args: (neg_a, A, neg_b, B, c_mod, C, reuse_a, reuse_b)
        acc = __builtin_amdgcn_wmma_f32_16x16x4_f32(
                  false, a, false, bm, (short)0, acc, false, false);
    }

    // ---- scale by 1/(|i||j|), ReLU, store ----
    const int ncol = tj * 16 + (lane & 15);
    const float sn = scl[ncol];
    float* Ob = O + (size_t)b * TILE_N;
    #pragma unroll
    for (int v = 0; v < 8; ++v) {
        const int mr = ti * 16 + v + 8 * (lane >> 4);
        const float val = acc[v] * scl[mr] * sn;
        Ob[mr * 32 + ncol] = fmaxf(val, 0.f);
    }
}

// ---------------------------------------------------------------------------
// Kernel 0: zero the global-max slot in workspace.
// ---------------------------------------------------------------------------
__global__ void init_max_kernel(unsigned* gmax) { *gmax = 0u; }

// ---------------------------------------------------------------------------
// Kernel 2: elementwise power fusion + ReLU, write raw fused; block max ->
// atomicMax on float-as-uint (post-ReLU values are >= 0, ordering preserved).
// fused[b,i,j] = relu( sum_{k,l} Pv[k] * a_param[k,l] * Pt[l] ),
// Pv[0]=Pt[0]=eye(i,j), Pv[k]=av^k, Pt[l]=at^l (elementwise powers).
// ---------------------------------------------------------------------------
__global__ void __launch_bounds__(256)
fuse_kernel(const float* __restrict__ av, const float* __restrict__ at,
            const float* __restrict__ a_param, float* __restrict__ fused,
            unsigned* __restrict__ gmax)
{
    __shared__ float sP[64];
    __shared__ float smax[8];
    const int tid = threadIdx.x;
    if (tid < 64) sP[tid] = a_param[tid];
    __syncthreads();

    const unsigned idx = blockIdx.x * 256u + tid;
    float result = 0.f;
    if (idx < TOTAL) {
        const int within = idx & (TILE_N - 1);
        const int i = within >> 5, j = within & 31;
        const float e   = (i == j) ? 1.f : 0.f;
        const float avv = av[idx];
        const float att = at[idx];

        float tp[8];
        tp[0] = e;
        float p = 1.f;
        #pragma unroll
        for (int l = 1; l < 8; ++l) { p *= att; tp[l] = p; }

        float acc = 0.f;
        float vp = 1.f;
        #pragma unroll
        for (int k = 0; k < 8; ++k) {
            float pk;
            if (k == 0) pk = e;
            else { vp *= avv; pk = vp; }
            float t = 0.f;
            #pragma unroll
            for (int l = 0; l < 8; ++l) t = fmaf(sP[k * 8 + l], tp[l], t);
            acc = fmaf(pk, t, acc);
        }
        result = fmaxf(acc, 0.f);
        fused[idx] = result;
    }

    // block max reduction
    #pragma unroll
    for (int m = 16; m >= 1; m >>= 1)
        result = fmaxf(result, __shfl_xor(result, m, 32));
    if ((tid & 31) == 0) smax[tid >> 5] = result;
    __syncthreads();
    if (tid < 8) {
        float r = smax[tid];
        #pragma unroll
        for (int m = 4; m >= 1; m >>= 1) r = fmaxf(r, __shfl_xor(r, m, 8));
        if (tid == 0) atomicMax(gmax, __float_as_uint(r));
    }
}

// ---------------------------------------------------------------------------
// Kernel 3: scale fused in place by 1/(max+0.01); copy a_param to output tail.
// ---------------------------------------------------------------------------
__global__ void __launch_bounds__(256)
finalize_kernel(float* __restrict__ fused, const unsigned* __restrict__ gmax,
                const float* __restrict__ a_param, float* __restrict__ out_param)
{
    const float inv = 1.0f / (__uint_as_float(*gmax) + 0.01f);
    const unsigned idx = blockIdx.x * 256u + threadIdx.x;
    if (idx < TOTAL) fused[idx] *= inv;
    if (blockIdx.x == 0 && threadIdx.x < 64)
        out_param[threadIdx.x] = a_param[threadIdx.x];
}

// ---------------------------------------------------------------------------
extern "C" void kernel_launch(void* const* d_in, const int* in_sizes, int n_in,
                              void* d_out, int out_size, void* d_ws, size_t ws_size,
                              hipStream_t stream)
{
    const float* fv      = (const float*)d_in[0];
    const float* ft      = (const float*)d_in[1];
    const float* a_param = (const float*)d_in[2];

    float* out   = (float*)d_out;
    float* o_av    = out;                       // B*S*S
    float* o_at    = out + (size_t)TOTAL;       // B*S*S
    float* o_fused = out + (size_t)2 * TOTAL;   // B*S*S
    float* o_param = out + (size_t)3 * TOTAL;   // 64

    unsigned* gmax = (unsigned*)d_ws;           // 4 bytes of workspace

    init_max_kernel<<<1, 1, 0, stream>>>(gmax);

    gram_wmma_kernel<<<BB, 256, 0, stream>>>(fv, ft, o_av, o_at);

    const unsigned nblk = (TOTAL + 255u) / 256u;   // 8192
    fuse_kernel<<<nblk, 256, 0, stream>>>(o_av, o_at, a_param, o_fused, gmax);

    finalize_kernel<<<nblk, 256, 0, stream>>>(o_fused, gmax, a_param, o_param);
}